// SO33Activation_12902081757699
// MI455X (gfx1250) — compile-verified
//
#include <hip/hip_runtime.h>
#include <hip/hip_bf16.h>
#include <math.h>

typedef __attribute__((ext_vector_type(2))) float v2f;
typedef __attribute__((ext_vector_type(8))) float v8f;

#define NB 15
#define DD 6

// ---------------------------------------------------------------------------
// Setup: omega = sum coeffs[k]*basis[k]; scale = 1/(1+||omega||_F);
// M = expm(scale*omega) via Taylor (||scale*omega||_F < 1 guaranteed).
// Emit WMMA A-operand lane tables for Mpad (16x8, two K=4 steps) into ws:
//   ws[0..31]   = A-step0 VGPR0 per lane  (K=0 | K=2)
//   ws[32..63]  = A-step0 VGPR1 per lane  (K=1 | K=3)
//   ws[64..95]  = A-step1 VGPR0 per lane  (K=4 | K=6->0)
//   ws[96..127] = A-step1 VGPR1 per lane  (K=5 | K=7->0)
// ---------------------------------------------------------------------------
__global__ void so33_prepare(const float* __restrict__ coeffs,
                             const float* __restrict__ basis,
                             float* __restrict__ ws)
{
    if (threadIdx.x != 0 || blockIdx.x != 0) return;

    float om[DD * DD];
    for (int e = 0; e < DD * DD; ++e) {
        float s = 0.f;
        for (int k = 0; k < NB; ++k) s += coeffs[k] * basis[k * DD * DD + e];
        om[e] = s;
    }
    float fr2 = 0.f;
    for (int e = 0; e < DD * DD; ++e) fr2 += om[e] * om[e];
    float frob  = sqrtf(fr2);
    float scale = 1.0f / (1.0f + frob);   // T = 1.0

    float A[DD * DD];
    for (int e = 0; e < DD * DD; ++e) A[e] = scale * om[e];

    // expm: Taylor series, ||A||_F = frob/(1+frob) < 1 -> 16 terms >> f32 eps
    float M[DD * DD], term[DD * DD], nxt[DD * DD];
    for (int e = 0; e < DD * DD; ++e) { M[e] = 0.f; term[e] = 0.f; }
    for (int i = 0; i < DD; ++i) { M[i * DD + i] = 1.f; term[i * DD + i] = 1.f; }
    for (int k = 1; k <= 16; ++k) {
        float invk = 1.0f / (float)k;
        for (int i = 0; i < DD; ++i)
            for (int j = 0; j < DD; ++j) {
                float s = 0.f;
                for (int p = 0; p < DD; ++p) s += term[i * DD + p] * A[p * DD + j];
                nxt[i * DD + j] = s * invk;
            }
        for (int e = 0; e < DD * DD; ++e) { term[e] = nxt[e]; M[e] += term[e]; }
    }

    // A-operand lane tables (ISA 32-bit A 16x4 layout: lanes0-15 K, lanes16-31 K+2)
    for (int lane = 0; lane < 32; ++lane) {
        int j = lane & 15, half = lane >> 4;
        float m0 = 0.f, m1 = 0.f, m2 = 0.f, m3 = 0.f;
        if (j < DD) {
            int k0 = half ? 2 : 0, k1 = half ? 3 : 1;
            m0 = M[j * DD + k0];
            m1 = M[j * DD + k1];
            if (!half) { m2 = M[j * DD + 4]; m3 = M[j * DD + 5]; }  // K=6,7 pad->0
        }
        ws[lane]      = m0;
        ws[32 + lane] = m1;
        ws[64 + lane] = m2;
        ws[96 + lane] = m3;
    }
}

// ---------------------------------------------------------------------------
// Main: one wave32 per 16-row tile.  D = Mpad(16x4) x x^T(4x16) (+ K=4..7 step)
// => lane n (n<16) holds y[row_base+n, 0..5] in C[0..5].  Streaming NT policy.
// ---------------------------------------------------------------------------
__global__ __launch_bounds__(256) void so33_apply(const float* __restrict__ x,
                                                  const float* __restrict__ aop,
                                                  float* __restrict__ y,
                                                  int nrows)
{
    const int lane = threadIdx.x & 31;
    const int wave = blockIdx.x * (blockDim.x >> 5) + (threadIdx.x >> 5);
    const long row_base = (long)wave * 16;
    if (row_base >= nrows) return;          // uniform per wave: EXEC stays all-1s

    const int  n    = lane & 15;
    const int  half = lane >> 4;
    long row = row_base + n;
    long rld = row < nrows ? row : (long)nrows - 1;   // clamp loads, keep EXEC full

    // A operands (uniform table, stays hot in cache)
    v2f A0 = { aop[lane],      aop[32 + lane] };
    v2f A1 = { aop[64 + lane], aop[96 + lane] };

    // B operands: 4x16 layout -> VGPR0 = K0|K2, VGPR1 = K1|K3 per lane half
    const float* xr = x + rld * 6;
    v2f B0 = __builtin_nontemporal_load((const v2f*)(xr + (half ? 2 : 0)));
    v2f B1 = { 0.f, 0.f };
    if (!half) B1 = __builtin_nontemporal_load((const v2f*)(xr + 4));

    v8f C = {};
    C = __builtin_amdgcn_wmma_f32_16x16x4_f32(false, A0, false, B0,
                                              (short)0, C, false, false);
    C = __builtin_amdgcn_wmma_f32_16x16x4_f32(false, A1, false, B1,
                                              (short)0, C, false, false);

    // C/D layout: lane n (<16), VGPR r => D[M=r, N=n] = y[row, r]
    if (!half && row < nrows) {
        float* yr = y + row * 6;
        v2f o0 = { C[0], C[1] };
        v2f o1 = { C[2], C[3] };
        v2f o2 = { C[4], C[5] };
        __builtin_nontemporal_store(o0, (v2f*)(yr));
        __builtin_nontemporal_store(o1, (v2f*)(yr + 2));
        __builtin_nontemporal_store(o2, (v2f*)(yr + 4));
    }
}

// ---------------------------------------------------------------------------
extern "C" void kernel_launch(void* const* d_in, const int* in_sizes, int n_in,
                              void* d_out, int out_size, void* d_ws, size_t ws_size,
                              hipStream_t stream)
{
    const float* x      = (const float*)d_in[0];   // (B, 6) f32
    const float* coeffs = (const float*)d_in[1];   // (15,)  f32
    const float* basis  = (const float*)d_in[2];   // (15,6,6) f32
    float*       out    = (float*)d_out;           // (B, 6) f32
    float*       ws     = (float*)d_ws;

    const int nrows = in_sizes[0] / 6;

    so33_prepare<<<1, 32, 0, stream>>>(coeffs, basis, ws);

    const int threads = 256;                 // 8 waves -> 128 rows per block
    const int waves   = (nrows + 15) / 16;
    const int wpb     = threads / 32;
    const int blocks  = (waves + wpb - 1) / wpb;
    so33_apply<<<blocks, threads, 0, stream>>>(x, ws, out, nrows);
}